// BilinearInterpolation_37546604102039
// MI455X (gfx1250) — compile-verified
//
#include <hip/hip_runtime.h>
#include <hip/hip_bf16.h>
#include <math.h>

// Problem constants (match the JAX reference)
#define BATCH 32
#define IH 512
#define IW 512
#define CH 3
#define OH 224
#define OW 224
#define NJ (OW * CH)             // 672 flattened (ox, c) columns
#define SCALE 0.4375f            // 224/512 (resize scale)
#define INVS 2.2857142857142856f // 512/224 = triangle kernel radius (antialias)

typedef __attribute__((ext_vector_type(2))) float v2f;
typedef __attribute__((ext_vector_type(8))) float v8f;

// ---------------------------------------------------------------------------
// Pass A: affine warp (bilinear, zero pad) + antialiased horizontal triangle
// resample, staged through LDS. One workgroup per (b, y) image row:
//   stage 1: 256 threads warp the 512 source pixels of row y into LDS
//            (each warped pixel = 4 corner gathers; input is L2-resident)
//   stage 2: 224 threads resample the LDS row to 224 output columns
// This removes the ~2.2x redundant gather work of per-output-column warping.
// ---------------------------------------------------------------------------
__global__ __launch_bounds__(256) void warp_hresize_lds_kernel(
    const float* __restrict__ img, const float* __restrict__ mats,
    float* __restrict__ tmp) {
  __shared__ float row[IW * CH];  // 6 KB warped-row staging buffer
  const int tid = threadIdx.x;
  const int y = blockIdx.x % IH;
  const int b = blockIdx.x / IH;
  const float* M = mats + b * 6;              // scalar (uniform) loads
  const float m00 = M[0], m01 = M[1], m02 = M[2];
  const float m10 = M[3], m11 = M[4], m12 = M[5];
  const float* imgb = img + (size_t)b * IH * IW * CH;
  const float fy = (float)y;
  const float byc = fmaf(m00, fy, m02);       // y-dependent affine part, hoisted
  const float bxc = fmaf(m10, fy, m12);

#pragma unroll
  for (int i = 0; i < 2; ++i) {
    const int x = tid + i * 256;
    const float fx = (float)x;
    const float sy = fmaf(m01, fx, byc);
    const float sx = fmaf(m11, fx, bxc);
    const float yf = floorf(sy), xf = floorf(sx);
    const float wy = sy - yf, wx = sx - xf;
    const int y0 = (int)yf, x0 = (int)xf;
    float v[2][2][3];
#pragma unroll
    for (int dy = 0; dy < 2; ++dy) {
#pragma unroll
      for (int dx = 0; dx < 2; ++dx) {
        const int yi = y0 + dy, xi = x0 + dx;
        const bool ok =
            ((unsigned)yi < (unsigned)IH) & ((unsigned)xi < (unsigned)IW);
        // Branch-free zero-pad: read a safe address, mask the value.
        const float* p =
            imgb + ((size_t)(ok ? yi : 0) * IW + (ok ? xi : 0)) * CH;
        const float s = ok ? 1.f : 0.f;
        v[dy][dx][0] = p[0] * s;
        v[dy][dx][1] = p[1] * s;
        v[dy][dx][2] = p[2] * s;
      }
    }
#pragma unroll
    for (int c = 0; c < 3; ++c) {
      const float top = fmaf(v[0][1][c] - v[0][0][c], wx, v[0][0][c]);
      const float bot = fmaf(v[1][1][c] - v[1][0][c], wx, v[1][0][c]);
      row[x * CH + c] = fmaf(bot - top, wy, top);
    }
  }
  __syncthreads();

  const int ox = tid;
  if (ox < OW) {
    const float pos = (ox + 0.5f) * INVS - 0.5f;
    int xlo = (int)ceilf(pos - INVS);  if (xlo < 0) xlo = 0;
    int xhi = (int)floorf(pos + INVS); if (xhi > IW - 1) xhi = IW - 1;
    float a0 = 0.f, a1 = 0.f, a2 = 0.f, ws = 0.f;
    for (int x = xlo; x <= xhi; ++x) {
      const float w = 1.f - fabsf((float)x - pos) * SCALE;
      if (w <= 0.f) continue;
      ws += w;
      a0 = fmaf(w, row[x * CH + 0], a0);
      a1 = fmaf(w, row[x * CH + 1], a1);
      a2 = fmaf(w, row[x * CH + 2], a2);
    }
    const float inv = 1.f / ws;
    float* dst = tmp + ((size_t)(b * IH + y) * NJ + ox * CH);
    dst[0] = a0 * inv; dst[1] = a1 * inv; dst[2] = a2 * inv;
  }
}

// ---------------------------------------------------------------------------
// Pass B: vertical resize as banded GEMM on the matrix pipe.
//   out[b](224 x 672) = Wv(224 x 512, banded triangle) * tmp[b](512 x 672)
// One wave per 16x16 tile; K iterated only over the ~42-row band, unrolled
// 2x (8 source rows / iteration) with V_WMMA_F32_16X16X4_F32. Tail tiles are
// harmless: triangle weight at distance > INVS is exactly 0 (1-INVS*SCALE==0)
// and row reads are clamped, so over-read tiles contribute 0*finite.
//
// VGPR layouts (ISA 7.12.2, f32):
//   A 16x4 : lane&15 = M; lane<16 -> {K=k0,k0+1}, lane>=16 -> {K=k0+2,k0+3}
//   B 4x16 : lane&15 = N; same K split as A across the two halves
//   C/D    : VGPR v = row m0 + v + 8*(lane>>4), col n0 + (lane&15)
// ---------------------------------------------------------------------------
__global__ __launch_bounds__(32) void vresize_wmma_kernel(
    const float* __restrict__ tmp, float* __restrict__ out) {
  const int n0   = blockIdx.x * 16;   // 0..656
  const int m0   = blockIdx.y * 16;   // 0..208
  const int b    = blockIdx.z;
  const int lane = threadIdx.x;
  const int half = lane >> 4;
  const int l16  = lane & 15;
  const int n    = n0 + l16;

  // Per-lane output row (A-matrix row): normalization over in-range taps.
  const int m = m0 + l16;
  const float posm = (m + 0.5f) * INVS - 0.5f;
  int ylo = (int)ceilf(posm - INVS);  if (ylo < 0) ylo = 0;
  int yhi = (int)floorf(posm + INVS); if (yhi > IH - 1) yhi = IH - 1;
  float rs = 0.f;
  for (int yy = ylo; yy <= yhi; ++yy) {
    const float w = 1.f - fabsf((float)yy - posm) * SCALE;
    rs += (w > 0.f) ? w : 0.f;
  }
  const float invrs = 1.f / rs;

  // Wave-uniform K band for this 16-row output tile (EXEC stays all-ones).
  const float plo = (m0 + 0.5f)  * INVS - 0.5f - INVS;
  const float phi = (m0 + 15.5f) * INVS - 0.5f + INVS;
  int kmin = (int)floorf(plo); if (kmin < 0) kmin = 0;
  int kmax = (int)ceilf(phi);  if (kmax > IH - 1) kmax = IH - 1;

  v8f acc = {};
  const float* tb = tmp + (size_t)b * IH * NJ;
  for (int k0 = kmin; k0 <= kmax; k0 += 8) {
#pragma unroll
    for (int u = 0; u < 2; ++u) {
      const int ka = k0 + 4 * u + 2 * half;  // the two K rows this lane supplies
      v2f a;
      {
        const float w0 = 1.f - fabsf((float)ka       - posm) * SCALE;
        const float w1 = 1.f - fabsf((float)(ka + 1) - posm) * SCALE;
        a.x = (ka     < IH && w0 > 0.f) ? w0 * invrs : 0.f;
        a.y = (ka + 1 < IH && w1 > 0.f) ? w1 * invrs : 0.f;
      }
      const int r0 = (ka     < IH) ? ka       : IH - 1;  // clamped; weight 0 there
      const int r1 = (ka + 1 < IH) ? (ka + 1) : IH - 1;
      v2f bv;
      bv.x = tb[(size_t)r0 * NJ + n];
      bv.y = tb[(size_t)r1 * NJ + n];
      acc = __builtin_amdgcn_wmma_f32_16x16x4_f32(false, a, false, bv,
                                                  (short)0, acc, false, false);
    }
  }

  // D layout: VGPR v holds output row m0 + v + 8*half at column n.
  // Output is streamed and never re-read -> non-temporal stores.
  float* ob = out + (size_t)b * OH * NJ;
#pragma unroll
  for (int v = 0; v < 8; ++v) {
    __builtin_nontemporal_store(acc[v], &ob[(size_t)(m0 + v + 8 * half) * NJ + n]);
  }
}

// ---------------------------------------------------------------------------
// Fallback (workspace too small): fully fused warp + 2D antialiased resize.
// ---------------------------------------------------------------------------
__device__ __forceinline__ void warp_hsample(const float* __restrict__ imgb,
                                             float m00, float m01, float m02,
                                             float m10, float m11, float m12,
                                             int y, int ox, float out3[3]) {
  const float pos = (ox + 0.5f) * INVS - 0.5f;
  int xlo = (int)ceilf(pos - INVS);  if (xlo < 0) xlo = 0;
  int xhi = (int)floorf(pos + INVS); if (xhi > IW - 1) xhi = IW - 1;
  float a0 = 0.f, a1 = 0.f, a2 = 0.f, wsum = 0.f;
  const float fy = (float)y;
  const float byc = fmaf(m00, fy, m02);
  const float bxc = fmaf(m10, fy, m12);
  for (int x = xlo; x <= xhi; ++x) {
    const float w = 1.f - fabsf((float)x - pos) * SCALE;
    if (w <= 0.f) continue;
    wsum += w;
    const float fx = (float)x;
    const float sy = fmaf(m01, fx, byc);
    const float sx = fmaf(m11, fx, bxc);
    const float yf = floorf(sy), xf = floorf(sx);
    const float wy = sy - yf, wx = sx - xf;
    const int y0 = (int)yf, x0 = (int)xf;
    float v[2][2][3];
#pragma unroll
    for (int dy = 0; dy < 2; ++dy) {
#pragma unroll
      for (int dx = 0; dx < 2; ++dx) {
        const int yi = y0 + dy, xi = x0 + dx;
        const bool ok =
            ((unsigned)yi < (unsigned)IH) & ((unsigned)xi < (unsigned)IW);
        const float* p =
            imgb + ((size_t)(ok ? yi : 0) * IW + (ok ? xi : 0)) * CH;
        const float s = ok ? 1.f : 0.f;
        v[dy][dx][0] = p[0] * s;
        v[dy][dx][1] = p[1] * s;
        v[dy][dx][2] = p[2] * s;
      }
    }
#pragma unroll
    for (int c = 0; c < 3; ++c) {
      const float top = fmaf(v[0][1][c] - v[0][0][c], wx, v[0][0][c]);
      const float bot = fmaf(v[1][1][c] - v[1][0][c], wx, v[1][0][c]);
      const float val = fmaf(bot - top, wy, top);
      if (c == 0)      a0 = fmaf(w, val, a0);
      else if (c == 1) a1 = fmaf(w, val, a1);
      else             a2 = fmaf(w, val, a2);
    }
  }
  const float inv = 1.f / wsum;
  out3[0] = a0 * inv; out3[1] = a1 * inv; out3[2] = a2 * inv;
}

__global__ __launch_bounds__(256) void fused_fallback_kernel(
    const float* __restrict__ img, const float* __restrict__ mats,
    float* __restrict__ out) {
  const int idx = blockIdx.x * 256 + threadIdx.x;
  if (idx >= BATCH * OH * OW) return;
  const int ox = idx % OW;
  const int t  = idx / OW;
  const int oy = t % OH;
  const int b  = t / OH;
  const float* M    = mats + b * 6;
  const float* imgb = img + (size_t)b * IH * IW * CH;
  const float posy = (oy + 0.5f) * INVS - 0.5f;
  int ylo = (int)ceilf(posy - INVS);  if (ylo < 0) ylo = 0;
  int yhi = (int)floorf(posy + INVS); if (yhi > IH - 1) yhi = IH - 1;
  float a0 = 0.f, a1 = 0.f, a2 = 0.f, ws = 0.f;
  for (int y = ylo; y <= yhi; ++y) {
    const float w = 1.f - fabsf((float)y - posy) * SCALE;
    if (w <= 0.f) continue;
    ws += w;
    float r[3];
    warp_hsample(imgb, M[0], M[1], M[2], M[3], M[4], M[5], y, ox, r);
    a0 = fmaf(w, r[0], a0); a1 = fmaf(w, r[1], a1); a2 = fmaf(w, r[2], a2);
  }
  const float inv = 1.f / ws;
  float* dst = out + ((size_t)(b * OH + oy) * OW + ox) * CH;
  dst[0] = a0 * inv; dst[1] = a1 * inv; dst[2] = a2 * inv;
}

extern "C" void kernel_launch(void* const* d_in, const int* in_sizes, int n_in,
                              void* d_out, int out_size, void* d_ws, size_t ws_size,
                              hipStream_t stream) {
  const float* images = (const float*)d_in[0];   // (32, 512, 512, 3) f32
  const float* mats   = (const float*)d_in[1];   // (32, 2, 3) f32
  float* out = (float*)d_out;                    // (32, 224, 224, 3) f32

  const size_t tmp_bytes = (size_t)BATCH * IH * NJ * sizeof(float);  // 44 MB
  if (d_ws != nullptr && ws_size >= tmp_bytes) {
    float* tmp = (float*)d_ws;
    // Pass A: one workgroup per (b, y) row; 16384 blocks x 256 threads.
    warp_hresize_lds_kernel<<<BATCH * IH, 256, 0, stream>>>(images, mats, tmp);
    // Pass B: 42 x 14 x 32 WMMA tiles, one wave each.
    dim3 grid(NJ / 16, OH / 16, BATCH);
    vresize_wmma_kernel<<<grid, 32, 0, stream>>>(tmp, out);
  } else {
    const int totF = BATCH * OH * OW;
    fused_fallback_kernel<<<(totF + 255) / 256, 256, 0, stream>>>(images, mats, out);
  }
  (void)in_sizes; (void)n_in; (void)out_size;
}